// graphConv_77635828842937
// MI455X (gfx1250) — compile-verified
//
#include <hip/hip_runtime.h>

// Polynomial graph filter: out[b] = (sum_k W_k S^k) X_b
// Stage 1 (Horner, 7 launches):  T <- W_k + T @ S      (N x N GEMMs, ping-pong in d_ws)
// Stage 2 (batched apply):       out_b <- A @ X_b      (128 batches, N x D)
// All math in f32 via V_WMMA_F32_16X16X4_F32 (memory-bound problem; keep full precision).

typedef __attribute__((ext_vector_type(2))) float v2f;
typedef __attribute__((ext_vector_type(8))) float v8f;

#define TILE_M 64
#define TILE_N 64
#define TILE_K 32
#define LDA_S  36   // padded LDS row stride (floats) for A tile: 36*4B = 144B, 16B-aligned
#define LDB_S  68   // padded LDS row stride (floats) for B tile: 68*4B = 272B, 16B-aligned

__global__ __launch_bounds__(256)
void gemm_wmma_f32(const float* __restrict__ A,
                   const float* __restrict__ B,
                   const float* __restrict__ E,   // optional addend (W_k); may be null
                   float* __restrict__ Dst,
                   int K,
                   int lda, int ldb, int lde, int ldd,
                   long long strideB, long long strideD)
{
    __shared__ float As[TILE_M * LDA_S];
    __shared__ float Bs[TILE_K * LDB_S];

    const int m0 = blockIdx.x * TILE_M;
    const int n0 = blockIdx.z * TILE_N;
    const float* __restrict__ Bb = B + (size_t)blockIdx.y * (size_t)strideB;
    float* __restrict__ Db = Dst + (size_t)blockIdx.y * (size_t)strideD;

    const int t    = threadIdx.x;       // 0..255
    const int lane = t & 31;
    const int wid  = t >> 5;            // 0..7
    const int rm   = (wid & 3) * 16;    // wave row offset in 64x64 tile
    const int cn   = (wid >> 2) * 32;   // wave col offset in 64x64 tile
    const int Mi   = lane & 15;         // M (for A) / N (for B,C,D) index within 16
    const int hi   = lane >> 4;         // half-wave select
    const int ko2  = hi * 2;            // K sub-offset per ISA f32 16x16x4 layout

    v8f acc0 = {0.f, 0.f, 0.f, 0.f, 0.f, 0.f, 0.f, 0.f};
    v8f acc1 = {0.f, 0.f, 0.f, 0.f, 0.f, 0.f, 0.f, 0.f};

    for (int k0 = 0; k0 < K; k0 += TILE_K) {
        // ---- cooperative loads: A tile 64x32, B tile 32x64, float4 each ----
#pragma unroll
        for (int i = 0; i < 2; ++i) {
            int idx = t + i * 256;              // 0..511
            int row = idx >> 3;                 // 64 rows
            int col = (idx & 7) << 2;           // 8 float4 per row
            float4 v = *reinterpret_cast<const float4*>(
                A + (size_t)(m0 + row) * lda + (k0 + col));
            *reinterpret_cast<float4*>(&As[row * LDA_S + col]) = v;
        }
#pragma unroll
        for (int i = 0; i < 2; ++i) {
            int idx = t + i * 256;              // 0..511
            int row = idx >> 4;                 // 32 rows
            int col = (idx & 15) << 2;          // 16 float4 per row
            float4 v = *reinterpret_cast<const float4*>(
                Bb + (size_t)(k0 + row) * ldb + (n0 + col));
            *reinterpret_cast<float4*>(&Bs[row * LDB_S + col]) = v;
        }
        __syncthreads();

        // ---- 8 K-steps of 4, two 16x16 WMMA tiles per wave ----
#pragma unroll
        for (int k = 0; k < TILE_K; k += 4) {
            v2f a, b0, b1;
            const float* ap = &As[(rm + Mi) * LDA_S + k + ko2];
            a.x = ap[0];
            a.y = ap[1];
            b0.x = Bs[(k + ko2)     * LDB_S + cn + Mi];
            b0.y = Bs[(k + ko2 + 1) * LDB_S + cn + Mi];
            b1.x = Bs[(k + ko2)     * LDB_S + cn + 16 + Mi];
            b1.y = Bs[(k + ko2 + 1) * LDB_S + cn + 16 + Mi];
            acc0 = __builtin_amdgcn_wmma_f32_16x16x4_f32(
                false, a, false, b0, (short)0, acc0, false, false);
            acc1 = __builtin_amdgcn_wmma_f32_16x16x4_f32(
                false, a, false, b1, (short)0, acc1, false, false);
        }
        __syncthreads();
    }

    // ---- epilogue: optional E add, store per ISA C/D layout ----
#pragma unroll
    for (int r = 0; r < 8; ++r) {
        int row  = m0 + rm + hi * 8 + r;
        int col0 = n0 + cn + Mi;
        int col1 = col0 + 16;
        float v0 = acc0[r];
        float v1 = acc1[r];
        if (E) {
            v0 += E[(size_t)row * lde + col0];
            v1 += E[(size_t)row * lde + col1];
        }
        Db[(size_t)row * ldd + col0] = v0;
        Db[(size_t)row * ldd + col1] = v1;
    }
}

extern "C" void kernel_launch(void* const* d_in, const int* in_sizes, int n_in,
                              void* d_out, int out_size, void* d_ws, size_t ws_size,
                              hipStream_t stream) {
    (void)in_sizes; (void)n_in; (void)out_size; (void)ws_size;
    const int Bsz = 128, N = 1024, D = 64, NUM_K = 8;

    const float* nodes  = (const float*)d_in[0];   // [B, N, D]
    const float* weight = (const float*)d_in[1];   // [NUM_K, N, N]
    const float* gshift = (const float*)d_in[2];   // [N, N]
    float* out = (float*)d_out;                    // [B, N, D]

    float* t0 = (float*)d_ws;                      // N*N f32 ping
    float* t1 = t0 + (size_t)N * N;                // N*N f32 pong

    // T = W_{K-1}
    hipMemcpyAsync(t0, weight + (size_t)(NUM_K - 1) * N * N,
                   (size_t)N * N * sizeof(float),
                   hipMemcpyDeviceToDevice, stream);

    // Horner: T <- W_k + T @ S, k = K-2 .. 0
    float* cur = t0;
    float* nxt = t1;
    for (int k = NUM_K - 2; k >= 0; --k) {
        gemm_wmma_f32<<<dim3(N / TILE_M, 1, N / TILE_N), 256, 0, stream>>>(
            cur, gshift, weight + (size_t)k * N * N, nxt,
            /*K=*/N, /*lda=*/N, /*ldb=*/N, /*lde=*/N, /*ldd=*/N,
            /*strideB=*/0, /*strideD=*/0);
        float* tmp = cur; cur = nxt; nxt = tmp;
    }

    // out[b] = A @ X_b  (A = cur, X_b = nodes[b] as [N x D])
    gemm_wmma_f32<<<dim3(N / TILE_M, Bsz, 1), 256, 0, stream>>>(
        cur, nodes, nullptr, out,
        /*K=*/N, /*lda=*/N, /*ldb=*/D, /*lde=*/0, /*ldd=*/D,
        /*strideB=*/(long long)N * D, /*strideD=*/(long long)N * D);
}